// SocialGCNLayer_87866440942261
// MI455X (gfx1250) — compile-verified
//
#include <hip/hip_runtime.h>

typedef __attribute__((ext_vector_type(2))) float v2f;
typedef __attribute__((ext_vector_type(8))) float v8f;

#define EMBED_D 128
#define LDS_PITCH 130   // 520B row pitch -> conflict-free ds_load_b64 across a wave

// Guaranteed native non-returning FP32 atomic add (L2 atomic unit, STOREcnt).
__device__ __forceinline__ void atomic_add_f32_native(float* addr, float val) {
    asm volatile("global_atomic_add_f32 %0, %1, off"
                 :
                 : "v"(addr), "v"(val)
                 : "memory");
}

// ---------------------------------------------------------------------------
// Kernel 1: weighted = user_emb @ W  using V_WMMA_F32_16X16X4_F32 (full fp32).
// Block = 256 threads = 8 waves. Each block computes 16 rows x 128 cols.
// Each wave owns one 16x16 tile and accumulates over K=128 in steps of 4.
// ---------------------------------------------------------------------------
__global__ __launch_bounds__(256) void socialgcn_gemm_wmma_f32(
    const float* __restrict__ A,      // [N,128] user_emb
    const float* __restrict__ W,      // [128,128] social_weight (row-major)
    float* __restrict__ Cmat,         // [N,128] weighted (workspace)
    int N)
{
    __shared__ float ldsWT[EMBED_D * LDS_PITCH];   // W transposed: ldsWT[n*130 + k] = W[k][n]

    const int t = threadIdx.x;

    // Cooperative transposed stage of W into LDS (128*128 floats / 256 threads).
    #pragma unroll
    for (int i = 0; i < (EMBED_D * EMBED_D) / 256; ++i) {
        int idx = t + i * 256;
        int k = idx >> 7;            // row of W
        int n = idx & (EMBED_D - 1); // col of W
        ldsWT[n * LDS_PITCH + k] = W[idx];
    }
    __syncthreads();

    const int lane = t & 31;
    const int wave = t >> 5;
    const int half = lane >> 4;      // 0 -> K pair {k0,k0+1}, 1 -> {k0+2,k0+3}
    const int l15  = lane & 15;      // M (for A) / N (for B) within tile
    const int m0   = blockIdx.x * 16;
    const int n0   = wave * 16;

    // Clamp row for A loads (keeps EXEC all-ones for WMMA even on a ragged tail).
    int mrow = m0 + l15;
    if (mrow >= N) mrow = N - 1;

    const float* arow = A + (size_t)mrow * EMBED_D;          // this lane's A row
    const float* brow = ldsWT + (n0 + l15) * LDS_PITCH;      // this lane's B column (transposed)

    v8f c = {};
    #pragma unroll
    for (int k0 = 0; k0 < EMBED_D; k0 += 4) {
        // A frag (16x4 f32): lanes 0-15 -> {K=k0,k0+1}, lanes 16-31 -> {K=k0+2,k0+3}
        v2f a = *(const v2f*)(arow + k0 + 2 * half);
        // B frag (4x16 f32): mirrored K assignment, contiguous in transposed LDS
        v2f b = *(const v2f*)(brow + k0 + 2 * half);
        // 8 args: (neg_a, A, neg_b, B, c_mod, C, reuse_a, reuse_b)
        c = __builtin_amdgcn_wmma_f32_16x16x4_f32(false, a, false, b,
                                                  (short)0, c, false, false);
    }

    // C/D layout: VGPR r -> M = r (lanes 0-15) or r+8 (lanes 16-31); N = lane&15.
    const int mbase = m0 + half * 8;
    #pragma unroll
    for (int r = 0; r < 8; ++r) {
        int m = mbase + r;
        if (m < N)
            Cmat[(size_t)m * EMBED_D + n0 + l15] = c[r];
    }
}

// ---------------------------------------------------------------------------
// Kernel 2: out = user_emb  (residual init), vectorized float4 copy.
// ---------------------------------------------------------------------------
__global__ __launch_bounds__(256) void socialgcn_init_out(
    const float4* __restrict__ src, float4* __restrict__ dst, size_t n4)
{
    size_t i = (size_t)blockIdx.x * blockDim.x + threadIdx.x;
    if (i < n4) dst[i] = src[i];
}

// ---------------------------------------------------------------------------
// Kernel 3: out[rows[e]] += vals[e] * weighted[cols[e]]   (atomic scatter).
// One wave32 per edge; each lane handles 4 consecutive floats.
// weighted (25.6MB) stays L2-resident; edge lists are streamed non-temporally
// so they don't evict it; atomics execute in the L2 atomic units (no return).
// ---------------------------------------------------------------------------
__global__ __launch_bounds__(256) void socialgcn_edge_scatter(
    const float* __restrict__ weighted,
    const int* __restrict__ rows,
    const int* __restrict__ cols,
    const float* __restrict__ vals,
    float* __restrict__ out,
    int E)
{
    const int e    = blockIdx.x * 8 + (threadIdx.x >> 5);
    const int lane = threadIdx.x & 31;
    if (e >= E) return;

    // Edge lists are touched exactly once -> non-temporal loads.
    const int   row = __builtin_nontemporal_load(rows + e);
    const int   col = __builtin_nontemporal_load(cols + e);
    const float v   = __builtin_nontemporal_load(vals + e);

    const float4 g = ((const float4*)(weighted + (size_t)col * EMBED_D))[lane];
    float* dst = out + (size_t)row * EMBED_D + lane * 4;

    atomic_add_f32_native(dst + 0, v * g.x);
    atomic_add_f32_native(dst + 1, v * g.y);
    atomic_add_f32_native(dst + 2, v * g.z);
    atomic_add_f32_native(dst + 3, v * g.w);
}

// ---------------------------------------------------------------------------
extern "C" void kernel_launch(void* const* d_in, const int* in_sizes, int n_in,
                              void* d_out, int out_size, void* d_ws, size_t ws_size,
                              hipStream_t stream) {
    const float* user_emb = (const float*)d_in[0];   // [N,128]
    const float* W        = (const float*)d_in[1];   // [128,128]
    const int*   rows     = (const int*)d_in[2];     // [E]
    const int*   cols     = (const int*)d_in[3];     // [E]
    const float* vals     = (const float*)d_in[4];   // [E]
    float*       out      = (float*)d_out;           // [N,128]

    const int N = in_sizes[0] / EMBED_D;             // 50000
    const int E = in_sizes[2];                       // 800000

    float* weighted = (float*)d_ws;                  // N*128 floats scratch

    // 1) weighted = user_emb @ W  (WMMA f32)
    const int gemm_blocks = (N + 15) / 16;
    socialgcn_gemm_wmma_f32<<<gemm_blocks, 256, 0, stream>>>(user_emb, W, weighted, N);

    // 2) out = user_emb (residual)
    const size_t n4 = (size_t)N * EMBED_D / 4;
    socialgcn_init_out<<<(unsigned)((n4 + 255) / 256), 256, 0, stream>>>(
        (const float4*)user_emb, (float4*)out, n4);

    // 3) scatter: out[rows[e]] += vals[e] * weighted[cols[e]]
    socialgcn_edge_scatter<<<(E + 7) / 8, 256, 0, stream>>>(
        weighted, rows, cols, vals, out, E);
}